// SamplingEdgeConv_80685255623065
// MI455X (gfx1250) — compile-verified
//
#include <hip/hip_runtime.h>
#include <math.h>

typedef float v2f __attribute__((ext_vector_type(2)));
typedef float v8f __attribute__((ext_vector_type(8)));

#define BB 4
#define CC 64
#define NN 16384
#define KK 20
#define SS 10
#define OUTC 64
#define NPTS (BB * NN)            // 65536 points
#define ROWS (NPTS * SS)          // 655360 rows
#define TWOC 128

// ---------------- Kernel 0: transpose x (B,C,N) -> xt (B*N, C) ----------------
__global__ void k_transpose(const float* __restrict__ x, float* __restrict__ xt) {
    __shared__ float tile[32][65];
    int blk = blockIdx.x;
    int b = blk / (NN / 32);
    int n0 = (blk % (NN / 32)) * 32;
    int t = threadIdx.x; // 256
    #pragma unroll
    for (int cc = 0; cc < 8; ++cc) {
        int c = cc * 8 + (t >> 5);
        int nn = t & 31;
        tile[nn][c] = x[((size_t)(b * CC + c)) * NN + n0 + nn];
    }
    __syncthreads();
    #pragma unroll
    for (int i = 0; i < 8; ++i) {
        int c = t & 63;
        int nn = (t >> 6) + i * 4;
        xt[((size_t)(b * NN + n0 + nn)) * CC + c] = tile[nn][c];
    }
}

// ---------------- Kernel 1: top-S sample of K scores per point ----------------
__global__ void k_sample(const float* __restrict__ scores, const int* __restrict__ idxmat,
                         int* __restrict__ samp) {
    int p = blockIdx.x * blockDim.x + threadIdx.x;
    if (p >= NPTS) return;
    const float* sc = scores + (size_t)p * KK;
    const int* im = idxmat + (size_t)p * KK;
    float s[KK];
    int id[KK];
    #pragma unroll
    for (int k = 0; k < KK; ++k) { s[k] = sc[k]; id[k] = im[k]; }
    unsigned used = 0;
    #pragma unroll
    for (int t = 0; t < SS; ++t) {
        float best = -INFINITY;
        int bk = 0, bid = 0;
        #pragma unroll
        for (int k = 0; k < KK; ++k) {
            bool ok = ((used >> k) & 1u) == 0u;
            if (ok && s[k] > best) { best = s[k]; bk = k; bid = id[k]; }
        }
        used |= 1u << bk;
        samp[(size_t)p * SS + t] = bid;
    }
}

// ---------------- Kernel 2a: zero stats ----------------
__global__ void k_zero(float* __restrict__ stats) {
    if (threadIdx.x < 256) stats[threadIdx.x] = 0.0f;
}

// ---------------- Kernel 2b: per-channel sum / sumsq of dw ----------------
__global__ void k_stats(const float* __restrict__ xt, const int* __restrict__ samp,
                        const float* __restrict__ dww, float* __restrict__ stats) {
    int c = threadIdx.x; // 128 threads, one channel each
    float w = dww[c];
    const int rows_per_block = ROWS / 2048; // 320
    size_t r0 = (size_t)blockIdx.x * rows_per_block;
    size_t r1 = r0 + rows_per_block;
    float sm = 0.f, sq = 0.f;
    for (size_t r = r0; r < r1; ++r) {
        int p = (int)(r / SS);
        int j = samp[r];
        float v;
        if (c < CC)
            v = (xt[(size_t)j * CC + c] - xt[(size_t)p * CC + c]) * w;
        else
            v = xt[(size_t)p * CC + (c - CC)] * w;
        sm += v;
        sq += v * v;
    }
    atomicAdd(&stats[c], sm);
    atomicAdd(&stats[TWOC + c], sq);
}

// ---------------- Kernel 3: fold BN into per-channel affine ----------------
__global__ void k_finalize(const float* __restrict__ stats, const float* __restrict__ dww,
                           const float* __restrict__ gamma, const float* __restrict__ beta,
                           float* __restrict__ coefs) {
    int c = threadIdx.x;
    if (c >= TWOC) return;
    const float cnt = (float)ROWS;
    float mean = stats[c] / cnt;
    float var = stats[TWOC + c] / cnt - mean * mean;
    float inv = rsqrtf(var + 1e-5f);
    float sbn = gamma[c] * inv;
    coefs[c] = dww[c] * sbn;                 // A[c]
    coefs[TWOC + c] = beta[c] - mean * sbn;  // B[c]
}

// leaky(x) with slope 0.2 == max(x, 0.2x) since slope in (0,1)
__device__ __forceinline__ float leaky02(float x) { return fmaxf(x, 0.2f * x); }

// ---------------- Kernel 4: fused gather + affine + leaky + WMMA GEMM + max ----------------
// Block: 320 threads = 10 waves. Block handles 16 points (160 rows, 10 M-tiles of 16).
// A: fp32 16x4 fragments built on the fly; B: pw^T staged in LDS (pair-packed for b64 loads).
// K-loop split into two straight-line halves (no branches):
//   half 1 (c in [0,64)):  e = neigh - central   (cache central pairs in regs)
//   half 2 (c in [64,128)): e = central          (register-resident, no global loads)
__global__ __launch_bounds__(320) void k_main(const float* __restrict__ xt,
                                              const int* __restrict__ samp,
                                              const float* __restrict__ coefs,
                                              const float* __restrict__ pw,
                                              float* __restrict__ out) {
    __shared__ float smem[10560]; // phase1: pw[8192] + cA[128] + cB[128]; phase2: out stage [160*66]
    float* s_pw = smem;           // packed: [(c>>1)*128 + o*2 + (c&1)]
    float* s_cA = smem + 8192;
    float* s_cB = smem + 8320;

    int tid = threadIdx.x;
    for (int i = tid; i < OUTC * TWOC; i += 320) {
        int o = i >> 7, c = i & 127;
        s_pw[(c >> 1) * 128 + o * 2 + (c & 1)] = pw[i];
    }
    if (tid < TWOC) { s_cA[tid] = coefs[tid]; s_cB[tid] = coefs[TWOC + tid]; }
    __syncthreads();

    int w = tid >> 5, lane = tid & 31;
    int lm = lane & 15, hi = lane >> 4;
    int lr = w * 16 + lm;             // local row 0..159
    int pl = lr / SS, s = lr - pl * SS;
    size_t g = (size_t)blockIdx.x * 16 + pl; // global point
    int j = samp[g * SS + s];
    const float* ctr = xt + g * CC;
    const float* nbr = xt + (size_t)j * CC;

    v8f acc0 = {}, acc1 = {}, acc2 = {}, acc3 = {};
    v2f ctv[16]; // cached central pairs (channels 2*hi + {0,1} + 4*it)

    // ---- half 1: channels [0,64), e = neigh - central ----
    #pragma unroll
    for (int it = 0; it < 16; ++it) {
        const int kb = it * 4;
        const int c0 = kb + 2 * hi;
        v2f nb = *(const v2f*)(nbr + c0);
        v2f ct = *(const v2f*)(ctr + c0);
        ctv[it] = ct;
        float a0 = leaky02((nb[0] - ct[0]) * s_cA[c0] + s_cB[c0]);
        float a1 = leaky02((nb[1] - ct[1]) * s_cA[c0 + 1] + s_cB[c0 + 1]);
        v2f A; A[0] = a0; A[1] = a1;
        const float* bb = s_pw + (c0 >> 1) * 128 + lm * 2;
        v2f B0 = *(const v2f*)(bb + 0);
        v2f B1 = *(const v2f*)(bb + 32);
        v2f B2 = *(const v2f*)(bb + 64);
        v2f B3 = *(const v2f*)(bb + 96);
        acc0 = __builtin_amdgcn_wmma_f32_16x16x4_f32(false, A, false, B0, (short)0, acc0, false, false);
        acc1 = __builtin_amdgcn_wmma_f32_16x16x4_f32(false, A, false, B1, (short)0, acc1, false, false);
        acc2 = __builtin_amdgcn_wmma_f32_16x16x4_f32(false, A, false, B2, (short)0, acc2, false, false);
        acc3 = __builtin_amdgcn_wmma_f32_16x16x4_f32(false, A, false, B3, (short)0, acc3, false, false);
    }

    // ---- half 2: channels [64,128), e = central (register-resident) ----
    #pragma unroll
    for (int it = 0; it < 16; ++it) {
        const int kb = 64 + it * 4;
        const int c0 = kb + 2 * hi;
        v2f ct = ctv[it];
        float a0 = leaky02(ct[0] * s_cA[c0] + s_cB[c0]);
        float a1 = leaky02(ct[1] * s_cA[c0 + 1] + s_cB[c0 + 1]);
        v2f A; A[0] = a0; A[1] = a1;
        const float* bb = s_pw + (c0 >> 1) * 128 + lm * 2;
        v2f B0 = *(const v2f*)(bb + 0);
        v2f B1 = *(const v2f*)(bb + 32);
        v2f B2 = *(const v2f*)(bb + 64);
        v2f B3 = *(const v2f*)(bb + 96);
        acc0 = __builtin_amdgcn_wmma_f32_16x16x4_f32(false, A, false, B0, (short)0, acc0, false, false);
        acc1 = __builtin_amdgcn_wmma_f32_16x16x4_f32(false, A, false, B1, (short)0, acc1, false, false);
        acc2 = __builtin_amdgcn_wmma_f32_16x16x4_f32(false, A, false, B2, (short)0, acc2, false, false);
        acc3 = __builtin_amdgcn_wmma_f32_16x16x4_f32(false, A, false, B3, (short)0, acc3, false, false);
    }

    __syncthreads(); // everyone done reading s_pw/coefs; alias smem as output stage
    float* s_out = smem; // [row][66]
    #pragma unroll
    for (int r = 0; r < 8; ++r) {
        int row = w * 16 + r + hi * 8;
        float* dst = s_out + row * 66 + lm;
        dst[0]  = acc0[r];
        dst[16] = acc1[r];
        dst[32] = acc2[r];
        dst[48] = acc3[r];
    }
    __syncthreads();

    for (int item = tid; item < 16 * OUTC; item += 320) {
        int o = item >> 4, pl2 = item & 15;
        size_t gg = (size_t)blockIdx.x * 16 + pl2;
        int b = (int)(gg >> 14);
        int n = (int)(gg & (NN - 1));
        float m = -INFINITY;
        #pragma unroll
        for (int ss2 = 0; ss2 < SS; ++ss2)
            m = fmaxf(m, s_out[(pl2 * SS + ss2) * 66 + o]);
        out[(((size_t)b * OUTC + o) << 14) + n] = m;
    }
}

extern "C" void kernel_launch(void* const* d_in, const int* in_sizes, int n_in,
                              void* d_out, int out_size, void* d_ws, size_t ws_size,
                              hipStream_t stream) {
    (void)in_sizes; (void)n_in; (void)out_size; (void)ws_size;
    const float* x      = (const float*)d_in[0];
    const int*   idxf   = (const int*)d_in[1];
    const float* scores = (const float*)d_in[2];
    const float* dww    = (const float*)d_in[3];
    const float* pw     = (const float*)d_in[4];
    const float* gamma  = (const float*)d_in[5];
    const float* beta   = (const float*)d_in[6];
    float* out = (float*)d_out;

    char* ws = (char*)d_ws;
    float* xt    = (float*)ws;                                  // B*N*C floats = 16 MB
    int*   samp  = (int*)(ws + (size_t)NPTS * CC * 4);          // ROWS ints
    float* stats = (float*)(ws + (size_t)NPTS * CC * 4 + (size_t)ROWS * 4);
    float* coefs = stats + 256;

    k_transpose<<<BB * (NN / 32), 256, 0, stream>>>(x, xt);
    k_sample<<<(NPTS + 255) / 256, 256, 0, stream>>>(scores, idxf, samp);
    k_zero<<<1, 256, 0, stream>>>(stats);
    k_stats<<<2048, 128, 0, stream>>>(xt, samp, dww, stats);
    k_finalize<<<1, 128, 0, stream>>>(stats, dww, gamma, beta, coefs);
    k_main<<<NPTS / 16, 320, 0, stream>>>(xt, samp, coefs, pw, out);
}